// WindowAttention_27470610825456
// MI455X (gfx1250) — compile-verified
//
#include <hip/hip_runtime.h>
#include <hip/hip_bf16.h>

// Fully-fused SwinV2 window attention for gfx1250 (MI455X), fp32 WMMA path.
// One 256-thread (8-wave, wave32) workgroup per window; all intermediates in LDS.

typedef __attribute__((ext_vector_type(2))) float v2f;
typedef __attribute__((ext_vector_type(8))) float v8f;

#define WMMA_F32X4(a, b, c) \
  __builtin_amdgcn_wmma_f32_16x16x4_f32(false, (a), false, (b), (short)0, (c), false, false)

constexpr int NTOK  = 49;    // tokens per window
constexpr int NPAD  = 64;    // padded to 4 WMMA M-tiles
constexpr int DIM   = 192;
constexpr int HEADS = 6;
constexpr int HD    = 32;    // head dim
constexpr int QKVD  = 3 * DIM;   // 576
constexpr int NWIN  = 64;    // windows per image (mask leading dim)

constexpr int X_FLOATS   = NPAD * DIM;          // 12288 (also attention-output buffer)
constexpr int QKV_FLOATS = NPAD * QKVD;         // 36864
constexpr int P_FLOATS   = HEADS * NPAD * NPAD; // 24576
constexpr size_t SMEM_BYTES = (size_t)(X_FLOATS + QKV_FLOATS + P_FLOATS) * sizeof(float); // 288 KB

__global__ __launch_bounds__(256)
void swinv2_window_attn_fused(const float* __restrict__ x,
                              const float* __restrict__ mask,
                              const float* __restrict__ qkv_w,
                              const float* __restrict__ qkv_b,
                              const float* __restrict__ logit_scale,
                              const float* __restrict__ table,
                              const int*   __restrict__ rel_index,
                              const float* __restrict__ proj_w,
                              const float* __restrict__ proj_b,
                              float* __restrict__ out)
{
  extern __shared__ float smem[];
  float* xbuf = smem;                          // [64][192] x, later attention output
  float* qkv  = smem + X_FLOATS;               // [64][576] q|k|v rows
  float* pbuf = smem + X_FLOATS + QKV_FLOATS;  // [6][64][64] softmax probs

  const int tid  = threadIdx.x;
  const int lane = tid & 31;
  const int wave = tid >> 5;
  const int lm   = lane & 15;          // row/col within 16 (A/B/C lane mapping)
  const int khi  = (lane >> 4) << 1;   // K offset 0 or 2 for A/B operands
  const int rhi  = (lane >> 4) << 3;   // row offset 0 or 8 for C/D
  const int b    = blockIdx.x;
  const size_t xoff = (size_t)b * NTOK * DIM;

  // ---- phase 0: load x into LDS, zero pad rows of x and qkv ----
  {
    const float4* xg = (const float4*)(x + xoff);
    float4* xs = (float4*)xbuf;
    for (int i = tid; i < NTOK * DIM / 4; i += 256) xs[i] = xg[i];
    for (int i = tid; i < (NPAD - NTOK) * DIM;  i += 256) xbuf[NTOK * DIM  + i] = 0.f;
    for (int i = tid; i < (NPAD - NTOK) * QKVD; i += 256) qkv [NTOK * QKVD + i] = 0.f;
  }
  __syncthreads();

  // ---- phase 1: qkv[n,o] = sum_c x[n,c] * qkv_w[o,c] + qkv_b[o] ----
  // Waves own N-panels (16 outputs wide); 4 M-tile accumulators so each
  // weight panel is streamed from L2 exactly once per block.
  for (int nt = wave; nt < QKVD / 16; nt += 8) {
    v8f c0 = {}, c1 = {}, c2 = {}, c3 = {};
    const float* brow = qkv_w + (size_t)(nt * 16 + lm) * DIM + khi;
    const float* arow = xbuf + lm * DIM + khi;
    for (int k = 0; k < DIM; k += 4) {
      v2f bb = *(const v2f*)(brow + k);
      c0 = WMMA_F32X4(*(const v2f*)(arow + k),              bb, c0);
      c1 = WMMA_F32X4(*(const v2f*)(arow + 16 * DIM + k),   bb, c1);
      c2 = WMMA_F32X4(*(const v2f*)(arow + 32 * DIM + k),   bb, c2);
      c3 = WMMA_F32X4(*(const v2f*)(arow + 48 * DIM + k),   bb, c3);
    }
    const int col = nt * 16 + lm;
    const float bias = qkv_b[col];
#pragma unroll
    for (int i = 0; i < 8; ++i) {
      qkv[( 0 + rhi + i) * QKVD + col] = c0[i] + bias;
      qkv[(16 + rhi + i) * QKVD + col] = c1[i] + bias;
      qkv[(32 + rhi + i) * QKVD + col] = c2[i] + bias;
      qkv[(48 + rhi + i) * QKVD + col] = c3[i] + bias;
    }
  }
  __syncthreads();

  // ---- phase 2: L2-normalize q (fold in clipped logit scale) and k ----
  for (int u = tid; u < NTOK * HEADS * 2; u += 256) {
    const int n   = u % NTOK;
    const int r   = u / NTOK;
    const int h   = r % HEADS;
    const int isk = r / HEADS;
    float* p = qkv + n * QKVD + isk * DIM + h * HD;
    float s = 0.f;
    for (int d = 0; d < HD; ++d) s += p[d] * p[d];
    float sc = 1.0f / fmaxf(sqrtf(s), 1e-12f);
    if (!isk) sc *= expf(fminf(logit_scale[h], 4.605170185988091f)); // log(100)
    for (int d = 0; d < HD; ++d) p[d] *= sc;
  }
  __syncthreads();

  const int w = b & (NWIN - 1); // window index within image for mask
  // ---- phase 3/4: per (head, m-tile) unit: QK^T -> softmax -> P@V ----
  for (int u = wave; u < HEADS * 4; u += 8) {
    const int h  = u % HEADS;
    const int mt = u / HEADS;
    float* ph = pbuf + h * NPAD * NPAD;

    // QK^T: C[n,m] = sum_d qn[n,d] * kn[m,d]
    for (int nt = 0; nt < 4; ++nt) {
      v8f c = {};
      const float* qrow = qkv + (mt * 16 + lm) * QKVD + h * HD + khi;
      const float* krow = qkv + (nt * 16 + lm) * QKVD + DIM + h * HD + khi;
      for (int k = 0; k < HD; k += 4)
        c = WMMA_F32X4(*(const v2f*)(qrow + k), *(const v2f*)(krow + k), c);
      const int m = nt * 16 + lm;
#pragma unroll
      for (int i = 0; i < 8; ++i) {
        const int n = mt * 16 + rhi + i;
        float s;
        if (m >= NTOK)      s = -3.0e38f;                       // pad cols -> prob 0
        else if (n < NTOK)  s = c[i]
                              + table[rel_index[n * NTOK + m] * HEADS + h]
                              + mask[((size_t)w * NTOK + n) * NTOK + m];
        else                s = 0.f;                            // junk rows stay finite
        ph[n * NPAD + m] = s;
      }
    }
    __builtin_amdgcn_wave_barrier();

    // softmax: 2 lanes per row (16 rows), cross-half combine via shfl_xor(16)
    {
      float* prow = ph + (mt * 16 + lm) * NPAD + ((lane >> 4) << 5);
      float mx = -3.4e38f;
      for (int j = 0; j < 32; ++j) mx = fmaxf(mx, prow[j]);
      mx = fmaxf(mx, __shfl_xor(mx, 16));
      float sum = 0.f;
      for (int j = 0; j < 32; ++j) { float e = __expf(prow[j] - mx); prow[j] = e; sum += e; }
      sum += __shfl_xor(sum, 16);
      const float inv = 1.0f / sum;
      for (int j = 0; j < 32; ++j) prow[j] *= inv;
    }
    __builtin_amdgcn_wave_barrier();

    // P @ V: C[n,d] = sum_m p[n,m] * v[m,d]; output into recycled xbuf
    for (int nt = 0; nt < 2; ++nt) {
      v8f c = {};
      const float* arow = ph + (mt * 16 + lm) * NPAD + khi;
      const int d = h * HD + nt * 16 + lm;
      for (int k = 0; k < NPAD; k += 4) {
        v2f a = *(const v2f*)(arow + k);
        v2f bb;
        bb.x = qkv[(k + khi    ) * QKVD + 2 * DIM + d];
        bb.y = qkv[(k + khi + 1) * QKVD + 2 * DIM + d];
        c = WMMA_F32X4(a, bb, c);
      }
#pragma unroll
      for (int i = 0; i < 8; ++i)
        xbuf[(mt * 16 + rhi + i) * DIM + d] = c[i];
    }
  }
  __syncthreads();

  // ---- phase 5: out[n,o] = sum_c ao[n,c] * proj_w[o,c] + proj_b[o] ----
  for (int nt = wave; nt < DIM / 16; nt += 8) {
    v8f c0 = {}, c1 = {}, c2 = {}, c3 = {};
    const float* brow = proj_w + (size_t)(nt * 16 + lm) * DIM + khi;
    const float* arow = xbuf + lm * DIM + khi;
    for (int k = 0; k < DIM; k += 4) {
      v2f bb = *(const v2f*)(brow + k);
      c0 = WMMA_F32X4(*(const v2f*)(arow + k),            bb, c0);
      c1 = WMMA_F32X4(*(const v2f*)(arow + 16 * DIM + k), bb, c1);
      c2 = WMMA_F32X4(*(const v2f*)(arow + 32 * DIM + k), bb, c2);
      c3 = WMMA_F32X4(*(const v2f*)(arow + 48 * DIM + k), bb, c3);
    }
    const int col = nt * 16 + lm;
    const float pb = proj_b[col];
#pragma unroll
    for (int i = 0; i < 8; ++i) {
      int row = rhi + i;
      if (row < NTOK)      out[xoff + (row     ) * DIM + col] = c0[i] + pb;
      row = 16 + rhi + i;
      if (row < NTOK)      out[xoff + (row     ) * DIM + col] = c1[i] + pb;
      row = 32 + rhi + i;
      if (row < NTOK)      out[xoff + (row     ) * DIM + col] = c2[i] + pb;
      // rows 48..63 tile: only row 48 valid
      row = 48 + rhi + i;
      if (row < NTOK)      out[xoff + (row     ) * DIM + col] = c3[i] + pb;
    }
  }
}

extern "C" void kernel_launch(void* const* d_in, const int* in_sizes, int n_in,
                              void* d_out, int out_size, void* d_ws, size_t ws_size,
                              hipStream_t stream) {
  (void)in_sizes; (void)n_in; (void)out_size; (void)d_ws; (void)ws_size;
  const float* x           = (const float*)d_in[0];
  const float* mask        = (const float*)d_in[1];
  const float* qkv_w       = (const float*)d_in[2];
  const float* qkv_b       = (const float*)d_in[3];
  const float* logit_scale = (const float*)d_in[4];
  const float* table       = (const float*)d_in[5];
  const int*   rel_index   = (const int*)d_in[6];
  const float* proj_w      = (const float*)d_in[7];
  const float* proj_b      = (const float*)d_in[8];
  float* out = (float*)d_out;

  swinv2_window_attn_fused<<<dim3(4096), dim3(256), SMEM_BYTES, stream>>>(
      x, mask, qkv_w, qkv_b, logit_scale, table, rel_index, proj_w, proj_b, out);
}